// TopologicalSensor_4097398800794
// MI455X (gfx1250) — compile-verified
//
#include <hip/hip_runtime.h>
#include <math.h>

#define BATCH 8
#define TLEN 1600
#define DDIM 64
#define WIN 50
#define NW 32
#define NPTS 48
#define NPAIR 1128          // 48*47/2
#define SORTN 2048
#define NRAD 20
#define NBINS 20
#define NSER (BATCH*TLEN)   // 12800
#define NWINS (BATCH*NW)    // 256

typedef __attribute__((ext_vector_type(2))) float v2f;
typedef __attribute__((ext_vector_type(8))) float v8f;

// ---------------- kernel 1: series[b,t] = ||latent[b,t,:]|| (f32) ----------
__global__ __launch_bounds__(256) void series_kernel(const float* __restrict__ latent,
                                                     float* __restrict__ series) {
    int i = blockIdx.x * 256 + threadIdx.x;
    if (i >= NSER) return;
    const float4* p = (const float4*)(latent + (size_t)i * DDIM);
    float acc = 0.f;
#pragma unroll
    for (int q = 0; q < DDIM / 4; ++q) {
        float4 v = p[q];
        acc += v.x * v.x + v.y * v.y + v.z * v.z + v.w * v.w;
    }
    series[i] = sqrtf(acc);
}

// order-preserving double <-> u64 encoding for LDS atomic max
__device__ inline unsigned long long encD(double d) {
    unsigned long long u = (unsigned long long)__double_as_longlong(d);
    return (u & 0x8000000000000000ULL) ? ~u : (u | 0x8000000000000000ULL);
}
__device__ inline double decD(unsigned long long e) {
    unsigned long long u = (e & 0x8000000000000000ULL) ? (e ^ 0x8000000000000000ULL) : ~e;
    return __longlong_as_double((long long)u);
}

// ---------------- kernel 2: one workgroup per window ----------------------
__global__ __launch_bounds__(256) void window_kernel(const float* __restrict__ series,
                                                     float* __restrict__ feats) {
    const int tid = threadIdx.x;
    const int g   = blockIdx.x;
    const int base = (g / NW) * TLEN + (g % NW) * WIN;

    __shared__ float win[WIN];
    __shared__ float sq[NPTS];
    __shared__ float Dm[NPTS][NPTS];
    __shared__ float sv[SORTN];
    __shared__ double radii[NRAD];
    __shared__ float r_med_s;
    __shared__ int beta1[NRAD];
    __shared__ int parent[NRAD][NPTS];
    __shared__ float deaths[NPTS - 1];
    __shared__ float mind[NPTS];
    __shared__ int intree[NPTS];
    __shared__ double red_s[256], red_slog[256], red_maxl[256], red_minb[256], red_maxd[256];
    __shared__ int red_cnt[256], red_nev[256];
    __shared__ unsigned long long lamEnc[NBINS];
    __shared__ double H1[5];      // S, SLOG, MAXL, MINB, MAXD
    __shared__ int H1i[2];        // CNT_med, NEV
    __shared__ double h0f[4];     // beta0, ent0, maxp0, land0
    __shared__ int bmax_s;

    if (tid < WIN) win[tid] = series[base + tid];
    __syncthreads();
    if (tid < NPTS) {
        float a = win[tid], b = win[tid + 1], c = win[tid + 2];
        sq[tid] = a * a + b * b + c * c;
    }
    __syncthreads();
    // pairwise distances, reference f32 formula
    for (int idx = tid; idx < NPTS * NPTS; idx += 256) {
        int i = idx / NPTS, j = idx % NPTS;
        float dot = win[i] * win[j] + win[i + 1] * win[j + 1] + win[i + 2] * win[j + 2];
        float d2 = sq[i] + sq[j] - 2.0f * dot;
        Dm[i][j] = sqrtf(fmaxf(d2, 0.0f));
    }
    __syncthreads();
    // fill sort array with unique pair distances (dup array value at k = uniq[k>>1])
    for (int p = tid; p < SORTN; p += 256) {
        if (p < NPAIR) {
            int i = 0, off = 0;
            while (off + (NPTS - 1 - i) <= p) { off += NPTS - 1 - i; ++i; }
            int j = i + 1 + (p - off);
            sv[p] = Dm[i][j];
        } else {
            sv[p] = __builtin_huge_valf();
        }
    }
    __syncthreads();
    // bitonic sort of 2048 floats in LDS
    for (int k = 2; k <= SORTN; k <<= 1) {
        for (int j = k >> 1; j > 0; j >>= 1) {
            for (int t = tid; t < SORTN; t += 256) {
                int ixj = t ^ j;
                if (ixj > t) {
                    float a = sv[t], b = sv[ixj];
                    bool asc = ((t & k) == 0);
                    if ((a > b) == asc) { sv[t] = b; sv[ixj] = a; }
                }
            }
            __syncthreads();
        }
    }
    // percentile radii (f64, like jnp.percentile with f64 q) over duplicated array of n=2256
    if (tid < NRAD) {
        double q    = 10.0 + (80.0 * (double)tid) / 19.0;
        double fidx = q * 0.01 * (double)(2 * NPAIR - 1);
        double fl   = floor(fidx);
        int f       = (int)fl;
        double frac = fidx - fl;
        double vf = (double)sv[f >> 1];
        double vc = (double)sv[(f + 1) >> 1];
        radii[tid] = vf + frac * (vc - vf);
    }
    if (tid == 32) {
        // median stays f32 in the reference (weak-typed 0.5)
        float vf = sv[1127 >> 1];
        float vc = sv[1128 >> 1];
        r_med_s = vf + 0.5f * (vc - vf);
    }
    __syncthreads();

    // threads 0..19: union-find per radius (replaces f64 eigvalsh exactly)
    if (tid < NRAD) {
        int* par = parent[tid];
        for (int k = 0; k < NPTS; ++k) par[k] = k;
        double rad = radii[tid];
        int cnt = 0;
        for (int i = 0; i < NPTS - 1; ++i) {
            for (int j = i + 1; j < NPTS; ++j) {
                if ((double)Dm[i][j] <= rad) {
                    ++cnt;
                    int a = i; while (par[a] != a) { par[a] = par[par[a]]; a = par[a]; }
                    int b = j; while (par[b] != b) { par[b] = par[par[b]]; b = par[b]; }
                    if (a != b) { if (a < b) par[b] = a; else par[a] = b; }
                }
            }
        }
        int comps = 0;
        for (int k = 0; k < NPTS; ++k) {
            int a = k; while (par[a] != a) a = par[a];
            if (a == k) ++comps;
        }
        int b1 = cnt - NPTS + comps;
        beta1[tid] = b1 > 0 ? b1 : 0;
    }
    // thread 255 (different wave): Prim MST + all H0 features, overlapped with UF
    if (tid == 255) {
        for (int j = 0; j < NPTS; ++j) { mind[j] = Dm[0][j]; intree[j] = (j == 0); }
        for (int s = 0; s < NPTS - 1; ++s) {
            float best = 1e30f; int bj = 0;
            for (int j = 0; j < NPTS; ++j) {
                float v = intree[j] ? 1e30f : mind[j];
                if (v < best) { best = v; bj = j; }
            }
            deaths[s] = best; intree[bj] = 1;
            for (int j = 0; j < NPTS; ++j) mind[j] = fminf(mind[j], Dm[bj][j]);
        }
        double s0 = 0, slog0 = 0, mx = -1e300; int b0 = 0;
        float rm = r_med_s;
        for (int s = 0; s < NPTS - 1; ++s) {
            float d = deaths[s];
            if (d > rm) ++b0;
            double dd = (double)d;
            if (dd > 0) { s0 += dd; slog0 += dd * log(dd); }
            if (dd > mx) mx = dd;
        }
        double ent0 = (s0 > 0) ? (log(s0) - slog0 / s0) : 0.0;
        // landscape: births=0, t in [0, max death]
        double t1 = mx, lsum = 0;
        for (int k = 0; k < NBINS; ++k) {
            double t = t1 * (double)k / (double)(NBINS - 1);
            double lam = -1e300;
            for (int s = 0; s < NPTS - 1; ++s) {
                double tent = fmin(t, (double)deaths[s] - t);
                if (tent > lam) lam = tent;
            }
            if (lam < 0) lam = 0;
            lsum += lam;
        }
        h0f[0] = (double)b0; h0f[1] = ent0; h0f[2] = mx; h0f[3] = lsum / (double)NBINS;
    }
    __syncthreads();
    if (tid == 0) {
        int m = 0;
        for (int r = 0; r < NRAD; ++r) if (beta1[r] > m) m = beta1[r];
        bmax_s = m;
    }
    __syncthreads();

    // H1 persistence pass A: each thread walks ~5 levels across 20 radii
    {
        double s = 0, slog = 0, maxl = -1e300, minb = 1e300, maxd = -1e300;
        int cnt = 0, nev = 0;
        double rm = (double)r_med_s;
        int bmax = bmax_s;
        for (int lev = tid + 1; lev <= NPAIR; lev += 256) {
            if (lev > bmax) break;
            bool prev = false; int last = 0;
            for (int r = 0; r < NRAD; ++r) {
                bool cur = beta1[r] >= lev;
                if (cur && !prev) last = r;
                else if (!cur && prev) {
                    double bth = radii[last], dth = radii[r], life = dth - bth;
                    ++nev;
                    if (bth <= rm && rm < dth) ++cnt;
                    if (life > 0) { s += life; slog += life * log(life); }
                    if (life > maxl) maxl = life;
                    if (bth < minb) minb = bth;
                    if (dth > maxd) maxd = dth;
                }
                prev = cur;
            }
        }
        red_s[tid] = s; red_slog[tid] = slog; red_maxl[tid] = maxl;
        red_minb[tid] = minb; red_maxd[tid] = maxd;
        red_cnt[tid] = cnt; red_nev[tid] = nev;
    }
    __syncthreads();
    if (tid == 0) {
        double S = 0, SL = 0, ML = -1e300, MB = 1e300, MD = -1e300; int C = 0, NE = 0;
        for (int t = 0; t < 256; ++t) {
            S += red_s[t]; SL += red_slog[t];
            if (red_maxl[t] > ML) ML = red_maxl[t];
            if (red_minb[t] < MB) MB = red_minb[t];
            if (red_maxd[t] > MD) MD = red_maxd[t];
            C += red_cnt[t]; NE += red_nev[t];
        }
        H1[0] = S; H1[1] = SL; H1[2] = ML; H1[3] = MB; H1[4] = MD;
        H1i[0] = C; H1i[1] = NE;
    }
    if (tid < NBINS) lamEnc[tid] = encD(-1e300);
    __syncthreads();

    // pass B: landscape for H1 over [min birth, max death]
    if (H1i[1] > 0) {
        double t0 = H1[3], t1 = H1[4];
        int bmax = bmax_s;
        for (int lev = tid + 1; lev <= NPAIR; lev += 256) {
            if (lev > bmax) break;
            bool prev = false; int last = 0;
            for (int r = 0; r < NRAD; ++r) {
                bool cur = beta1[r] >= lev;
                if (cur && !prev) last = r;
                else if (!cur && prev) {
                    double bth = radii[last], dth = radii[r];
                    for (int k = 0; k < NBINS; ++k) {
                        double t = t0 + (t1 - t0) * (double)k / (double)(NBINS - 1);
                        double tent = fmin(t - bth, dth - t);
                        atomicMax(&lamEnc[k], encD(tent));
                    }
                }
                prev = cur;
            }
        }
    }
    __syncthreads();
    if (tid == 0) {
        double land1 = 0.0;
        if (H1i[1] > 0) {
            double ls = 0;
            for (int k = 0; k < NBINS; ++k) {
                double l = decD(lamEnc[k]);
                if (l < 0) l = 0;
                ls += l;
            }
            land1 = ls / (double)NBINS;
        }
        double S = H1[0];
        double ent1  = (S > 0) ? (log(S) - H1[1] / S) : 0.0;
        double maxp1 = (H1i[1] > 0) ? (H1[2] > 0 ? H1[2] : 0.0) : 0.0;
        float* o = feats + (size_t)g * 8;
        o[0] = (float)h0f[0];        // beta0
        o[1] = (float)H1i[0];        // beta1_med
        o[2] = (float)h0f[1];        // ent0
        o[3] = (float)ent1;          // ent1
        o[4] = (float)h0f[2];        // maxp0
        o[5] = (float)maxp1;         // maxp1
        o[6] = (float)h0f[3];        // land0
        o[7] = (float)land1;         // land1
    }
}

// ---------------- kernel 3: out = feats @ W^T + b via V_WMMA_F32_16X16X4_F32
// one wave per 16-row tile; K=8 split into two 16x16x4 f32 WMMAs.
__global__ __launch_bounds__(32) void proj_kernel(const float* __restrict__ feats,
                                                  const float* __restrict__ W,
                                                  const float* __restrict__ bias,
                                                  float* __restrict__ out) {
    const int tile = blockIdx.x;        // 0..15
    const int lane = threadIdx.x;       // wave32
    const int half = lane >> 4;         // 0: K base 0/2 rows M0..15 ; 1: K 2/3
    const int l    = lane & 15;
    const int row  = tile * 16 + l;     // A: M = l
    const int k0   = half * 2;

    // A 16x4 layout: lanes0-15 -> (K=0,K=1) in vgpr0/1; lanes16-31 -> (K=2,K=3)
    v2f a0, a1;
    a0.x = feats[row * 8 + k0];     a0.y = feats[row * 8 + k0 + 1];
    a1.x = feats[row * 8 + 4 + k0]; a1.y = feats[row * 8 + 5 + k0];

    // B 4x16 layout: vgpr0 holds rows K=0 (lanes0-15) / K=2 (lanes16-31), vgpr1 K=1/K=3
    // B[k][n] = W[n*8+k]; pad n>=8 with zeros
    float w0 = 0.f, w1 = 0.f, w2 = 0.f, w3 = 0.f;
    if (l < 8) {
        w0 = W[l * 8 + k0];     w1 = W[l * 8 + k0 + 1];
        w2 = W[l * 8 + 4 + k0]; w3 = W[l * 8 + 5 + k0];
    }
    v2f b0, b1;
    b0.x = w0; b0.y = w1;
    b1.x = w2; b1.y = w3;

    v8f c = {0.f, 0.f, 0.f, 0.f, 0.f, 0.f, 0.f, 0.f};
    c = __builtin_amdgcn_wmma_f32_16x16x4_f32(false, a0, false, b0, (short)0, c, false, false);
    c = __builtin_amdgcn_wmma_f32_16x16x4_f32(false, a1, false, b1, (short)0, c, false, false);

    // D layout: vgpr g -> row M=g (lanes0-15) / M=g+8 (lanes16-31), N=l
    if (l < 8) {
        float bb = bias[l];
#pragma unroll
        for (int g2 = 0; g2 < 8; ++g2) {
            out[(tile * 16 + g2 + half * 8) * 8 + l] = c[g2] + bb;
        }
    }
}

extern "C" void kernel_launch(void* const* d_in, const int* in_sizes, int n_in,
                              void* d_out, int out_size, void* d_ws, size_t ws_size,
                              hipStream_t stream) {
    const float* latent = (const float*)d_in[0];   // (8,1600,64) f32
    const float* W      = (const float*)d_in[1];   // (8,8) f32
    const float* b      = (const float*)d_in[2];   // (8,) f32
    float* out = (float*)d_out;                    // (8,32,8) f32

    float* series = (float*)d_ws;                  // 12800 f32
    float* feats  = series + NSER;                 // 256*8 f32

    series_kernel<<<(NSER + 255) / 256, 256, 0, stream>>>(latent, series);
    window_kernel<<<NWINS, 256, 0, stream>>>(series, feats);
    proj_kernel<<<16, 32, 0, stream>>>(feats, W, b, out);
}